// VectorQuantizer_32547262169387
// MI455X (gfx1250) — compile-verified
//
#include <hip/hip_runtime.h>
#include <hip/hip_bf16.h>

typedef __attribute__((ext_vector_type(2))) float v2f;
typedef __attribute__((ext_vector_type(8))) float v8f;

#define EMB_DIM   64
#define NUM_EMB   1024
#define ROWS_PER_WAVE 16
#define WAVES_PER_BLOCK 8
#define ROWS_PER_BLOCK (ROWS_PER_WAVE * WAVES_PER_BLOCK)   // 128
#define CODES_PER_ITER 32                                   // 2 column tiles
#define ITERS (NUM_EMB / CODES_PER_ITER)                    // 32

// ---------------------------------------------------------------------------
// Kernel 1: codebook squared norms -> d_ws (1024 floats)
// ---------------------------------------------------------------------------
__global__ __launch_bounds__(256) void vq_norms_kernel(
    const float* __restrict__ cb, float* __restrict__ norms) {
  int k = blockIdx.x * blockDim.x + threadIdx.x;
  if (k >= NUM_EMB) return;
  const float4* row = (const float4*)(cb + (size_t)k * EMB_DIM);
  float acc = 0.0f;
#pragma unroll
  for (int i = 0; i < EMB_DIM / 4; ++i) {
    float4 v = row[i];
    acc += v.x * v.x + v.y * v.y + v.z * v.z + v.w * v.w;
  }
  norms[k] = acc;
}

// ---------------------------------------------------------------------------
// Kernel 2: distances via f32 WMMA (4 independent accumulator chains),
// double-buffered LDS staging, per-row argmin, gather.
//   score = ||e||^2 - 2 z.e   (||z||^2 dropped; constant per row)
// ---------------------------------------------------------------------------
__global__ __launch_bounds__(256) void vq_main_kernel(
    const float* __restrict__ z, const float* __restrict__ cb,
    const float* __restrict__ norms, float* __restrict__ zq,
    int* __restrict__ idx_out) {
  __shared__ float lds_cb[2][CODES_PER_ITER * EMB_DIM];  // 2 x 8 KB
  __shared__ float lds_nm[2][CODES_PER_ITER];

  const int tid  = threadIdx.x;
  const int lane = tid & 31;
  const int wave = tid >> 5;
  const int row0 = (blockIdx.x * WAVES_PER_BLOCK + wave) * ROWS_PER_WAVE;

  const int n     = lane & 15;        // column within tile / M row of A tile
  const int khalf = (lane >> 4) * 2;  // K sub-offset for this half-wave

  // ---- load & pre-scale A fragments (live in VGPRs for the whole loop) ----
  const float* zrow = z + (size_t)(row0 + n) * EMB_DIM + khalf;
  v2f a[16];
#pragma unroll
  for (int f = 0; f < 16; ++f) {
    v2f t = *(const v2f*)(zrow + 4 * f);
    a[f] = t * -2.0f;
  }

  // ---- prologue: stage iteration 0 directly into buffer 0 ----
  {
    const float4* src = (const float4*)cb;
    float4* dst = (float4*)lds_cb[0];
    dst[tid]       = src[tid];
    dst[tid + 256] = src[tid + 256];
    if (tid < CODES_PER_ITER) lds_nm[0][tid] = norms[tid];
  }

  float bestV[8];
  int   bestI[8];
#pragma unroll
  for (int s = 0; s < 8; ++s) { bestV[s] = 3.0e38f; bestI[s] = 0; }

  for (int t = 0; t < ITERS; ++t) {
    const int cur = t & 1;
    const int nxt = cur ^ 1;
    __syncthreads();  // buffer[cur] published; buffer[nxt] fully consumed

    // ---- issue global loads for iteration t+1 (latency hidden by WMMAs) ---
    float4 r0, r1;
    float  rn = 0.0f;
    const bool have_next = (t + 1 < ITERS);
    if (have_next) {
      const float4* src =
          (const float4*)(cb + (size_t)(t + 1) * CODES_PER_ITER * EMB_DIM);
      r0 = src[tid];
      r1 = src[tid + 256];
      if (tid < CODES_PER_ITER) rn = norms[(t + 1) * CODES_PER_ITER + tid];
      if (t + 2 < ITERS)  // speculative prefetch of the tile after that
        __builtin_prefetch(
            cb + (size_t)(t + 2) * CODES_PER_ITER * EMB_DIM + tid * 8, 0, 0);
    }

    // ---- 32 WMMAs in 4 independent chains (2 col tiles x 2 K-halves) -----
    const float* L = lds_cb[cur];
    v8f c00 = {}, c01 = {}, c10 = {}, c11 = {};
#pragma unroll
    for (int f = 0; f < 8; ++f) {
      v2f b00 = *(const v2f*)(&L[n * EMB_DIM + 4 * f + khalf]);
      v2f b10 = *(const v2f*)(&L[(16 + n) * EMB_DIM + 4 * f + khalf]);
      v2f b01 = *(const v2f*)(&L[n * EMB_DIM + 4 * (f + 8) + khalf]);
      v2f b11 = *(const v2f*)(&L[(16 + n) * EMB_DIM + 4 * (f + 8) + khalf]);
      c00 = __builtin_amdgcn_wmma_f32_16x16x4_f32(false, a[f],     false, b00,
                                                  (short)0, c00, false, false);
      c10 = __builtin_amdgcn_wmma_f32_16x16x4_f32(false, a[f],     false, b10,
                                                  (short)0, c10, false, false);
      c01 = __builtin_amdgcn_wmma_f32_16x16x4_f32(false, a[f + 8], false, b01,
                                                  (short)0, c01, false, false);
      c11 = __builtin_amdgcn_wmma_f32_16x16x4_f32(false, a[f + 8], false, b11,
                                                  (short)0, c11, false, false);
    }
    const float ne0 = lds_nm[cur][n];
    const float ne1 = lds_nm[cur][16 + n];

    // ---- running argmin update (tile 0 first: lower indices win ties) ----
    const int kbase = t * CODES_PER_ITER + n;
    v8f d0 = c00 + c01;
    v8f d1 = c10 + c11;
#pragma unroll
    for (int s = 0; s < 8; ++s) {
      float v0 = ne0 + d0[s];
      if (v0 < bestV[s]) { bestV[s] = v0; bestI[s] = kbase; }
      float v1 = ne1 + d1[s];
      if (v1 < bestV[s]) { bestV[s] = v1; bestI[s] = kbase + 16; }
    }

    // ---- publish iteration t+1 into the other buffer ----------------------
    if (have_next) {
      float4* dst = (float4*)lds_cb[nxt];
      dst[tid]       = r0;
      dst[tid + 256] = r1;
      if (tid < CODES_PER_ITER) lds_nm[nxt][tid] = rn;
    }
  }

  // ---- argmin all-reduce across the 16 lanes holding one row ----
#pragma unroll
  for (int off = 1; off <= 8; off <<= 1) {
#pragma unroll
    for (int s = 0; s < 8; ++s) {
      float ov = __shfl_xor(bestV[s], off, 32);
      int   oi = __shfl_xor(bestI[s], off, 32);
      if (ov < bestV[s] || (ov == bestV[s] && oi < bestI[s])) {
        bestV[s] = ov; bestI[s] = oi;
      }
    }
  }
  // lanes 0-15 hold rows 0..7 (slot s), lanes 16-31 hold rows 8..15

  if (lane == 0) {
#pragma unroll
    for (int s = 0; s < 8; ++s) idx_out[row0 + s] = bestI[s];
  }
  if (lane == 16) {
#pragma unroll
    for (int s = 0; s < 8; ++s) idx_out[row0 + 8 + s] = bestI[s];
  }

  // ---- gather selected code rows: whole wave copies 64 floats per row ----
#pragma unroll
  for (int r = 0; r < 16; ++r) {
    int ridx = __shfl(bestI[r & 7], (r < 8) ? 0 : 16, 32);
    const v2f* src = (const v2f*)(cb + (size_t)ridx * EMB_DIM) + lane;
    v2f*       dst = (v2f*)(zq + (size_t)(row0 + r) * EMB_DIM) + lane;
    *dst = *src;
  }
}

extern "C" void kernel_launch(void* const* d_in, const int* in_sizes, int n_in,
                              void* d_out, int out_size, void* d_ws, size_t ws_size,
                              hipStream_t stream) {
  const float* z  = (const float*)d_in[0];
  const float* cb = (const float*)d_in[1];
  const int n_rows = in_sizes[0] / EMB_DIM;          // 65536

  float* norms   = (float*)d_ws;                     // 1024 floats scratch
  float* zq      = (float*)d_out;                    // output 0: z_q (fp32)
  int*   idx_out = (int*)((float*)d_out + (size_t)n_rows * EMB_DIM);  // output 1

  vq_norms_kernel<<<(NUM_EMB + 255) / 256, 256, 0, stream>>>(cb, norms);

  const int blocks = n_rows / ROWS_PER_BLOCK;        // 512
  vq_main_kernel<<<blocks, 256, 0, stream>>>(z, cb, norms, zq, idx_out);
}